// SoftmaxRouter_88089779241157
// MI455X (gfx1250) — compile-verified
//
#include <hip/hip_runtime.h>
#include <hip/hip_bf16.h>
#include <stdint.h>

typedef __attribute__((ext_vector_type(2))) float v2f;
typedef __attribute__((ext_vector_type(8))) float v8f;
typedef int v4i __attribute__((vector_size(16)));     // matches builtin param type

#define AS1 __attribute__((address_space(1)))
#define AS3 __attribute__((address_space(3)))

#define HIDDEN        2048
#define NEXP          64
#define KC            64                 // K-chunk staged in LDS
#define KSTEPS        (KC / 4)           // wmma K=4 steps per chunk
#define EXP_STRIDE    68                 // padded dwords per expert row in LDS (bank spread)
#define TOK_PER_WAVE  16
#define WAVES_PER_WG  8
#define TOK_PER_WG    (TOK_PER_WAVE * WAVES_PER_WG)   // 128
#define BUF_DW        (NEXP * EXP_STRIDE)             // 4352 dwords = 17408 B per buffer

#if __has_builtin(__builtin_amdgcn_global_load_async_to_lds_b128)
#define HAVE_ASYNC 1
#else
#define HAVE_ASYNC 0
#endif

__global__ __launch_bounds__(256) void SoftmaxRouter_kernel(
    const float* __restrict__ hs,   // [T, HIDDEN]
    const float* __restrict__ gw,   // [NEXP, HIDDEN]
    const float* __restrict__ gb,   // [NEXP]
    float* __restrict__ out,        // [T*2] weights ++ [T*2] indices (as float)
    int T)
{
    __shared__ float lds[2 * BUF_DW];   // 34816 B; reused for logit scratch

    const int tid     = threadIdx.x;
    const int lane    = tid & 31;
    const int wave    = tid >> 5;
    const int mLocal  = lane & 15;            // M index within 16-row tile
    const int half    = lane >> 4;            // 0: lanes 0-15, 1: lanes 16-31
    const int kPhase  = half << 1;            // K sub-offset 0 or 2 (A/B fragment layout)
    const int tokBase = blockIdx.x * TOK_PER_WG + wave * TOK_PER_WAVE;

    // ---- stage one 64 x KC slab of gate_w into LDS buffer `buf` ----
    auto copyB = [&](int kc, int buf) {
        const int base = buf * BUF_DW;
        #pragma unroll
        for (int i = 0; i < (NEXP * KC / 4) / 256; ++i) {   // 4 float4 per thread
            int j = i * 256 + tid;                // 0..1023 float4 tiles
            int e = j >> 4;                       // expert
            int q = j & 15;                       // float4 index within KC
            const float* gsrc = gw + e * HIDDEN + kc * KC + q * 4;
            float*       ldst = &lds[base + e * EXP_STRIDE + q * 4];
#if HAVE_ASYNC
            __builtin_amdgcn_global_load_async_to_lds_b128(
                (AS1 v4i*)gsrc, (AS3 v4i*)ldst, 0, 0);
#else
            *(float4*)ldst = *(const float4*)gsrc;
#endif
        }
    };
    auto waitAsync = [&]() {
#if HAVE_ASYNC
  #if __has_builtin(__builtin_amdgcn_s_wait_asynccnt)
        __builtin_amdgcn_s_wait_asynccnt(0);
  #else
        asm volatile("s_wait_asynccnt 0" ::: "memory");
  #endif
#endif
    };

    // ---- GEMM: 16 tokens x 64 experts per wave, fp32 WMMA 16x16x4 ----
    v8f c[4] = {};                                  // 4 N-tiles of 16 experts
    const float* aRow = hs + (size_t)(tokBase + mLocal) * HIDDEN;

    const int NCHUNK = HIDDEN / KC;                 // 32
    copyB(0, 0);
    waitAsync();
    __syncthreads();

    for (int kc = 0; kc < NCHUNK; ++kc) {
        if (kc + 1 < NCHUNK) copyB(kc + 1, (kc + 1) & 1);   // prefetch next slab
        const float* ldsB = &lds[(kc & 1) * BUF_DW];
        const float* aPtr = aRow + kc * KC + kPhase;

        #pragma unroll
        for (int s = 0; s < KSTEPS; ++s) {
            // A fragment: lanes 0-15 rows M=0..15 K={k,k+1}; lanes 16-31 K={k+2,k+3}
            v2f a = *(const v2f*)(aPtr + s * 4);
            #pragma unroll
            for (int n = 0; n < 4; ++n) {
                // B fragment: lane -> expert n*16+mLocal, K pair (kPhase, kPhase+1)
                v2f b = *(const v2f*)(ldsB + (n * 16 + mLocal) * EXP_STRIDE + s * 4 + kPhase);
                c[n] = __builtin_amdgcn_wmma_f32_16x16x4_f32(
                           false, a, false, b, (short)0, c[n], false, false);
            }
        }
        waitAsync();
        __syncthreads();
    }

    // ---- epilogue: bias, spill logits to LDS, per-token top-2 + softmax ----
    float* ldsL = &lds[wave * (TOK_PER_WAVE * 65)];     // 65-dword row stride: no bank conflicts
    #pragma unroll
    for (int n = 0; n < 4; ++n) {
        float bias = gb[n * 16 + mLocal];
        #pragma unroll
        for (int r = 0; r < 8; ++r) {
            int M = r + (half << 3);                     // C/D layout: lanes 16-31 hold M+8
            ldsL[M * 65 + n * 16 + mLocal] = c[n][r] + bias;
        }
    }
    // same-wave LDS RAW handled by DScnt waits the compiler inserts
    if (lane < 16) {
        const float* row = &ldsL[lane * 65];
        float b0 = -3.402823466e38f, b1 = -3.402823466e38f;
        int   i0 = 0, i1 = 0;
        for (int e = 0; e < NEXP; ++e) {
            float v = row[e];
            if (v > b0)      { b1 = b0; i1 = i0; b0 = v; i0 = e; }   // strict > => lowest idx on ties
            else if (v > b1) { b1 = v; i1 = e; }
        }
        float p1 = __expf(b1 - b0);
        float r  = 1.0f / (1.0f + p1);
        int tok = tokBase + lane;
        out[tok * 2 + 0] = r;            // softmax over [b0, b1]
        out[tok * 2 + 1] = p1 * r;
        float* oi = out + (size_t)T * 2;
        oi[tok * 2 + 0] = (float)i0;
        oi[tok * 2 + 1] = (float)i1;
    }
}

extern "C" void kernel_launch(void* const* d_in, const int* in_sizes, int n_in,
                              void* d_out, int out_size, void* d_ws, size_t ws_size,
                              hipStream_t stream) {
    const float* hs = (const float*)d_in[0];   // [4, 8192, 2048] fp32
    const float* gw = (const float*)d_in[1];   // [64, 2048] fp32
    const float* gb = (const float*)d_in[2];   // [64] fp32
    float* out = (float*)d_out;

    int T = in_sizes[0] / HIDDEN;              // 32768 tokens
    int grid = T / TOK_PER_WG;                 // 256 workgroups
    SoftmaxRouter_kernel<<<grid, 256, 0, stream>>>(hs, gw, gb, out, T);
}